// ODEFunc_9105330668117
// MI455X (gfx1250) — compile-verified
//
#include <hip/hip_runtime.h>
#include <hip/hip_bf16.h>

// ---- problem constants (from reference) ----
#define S_   4
#define N_   50000
#define E_   800000
#define DZ_  32
#define DH_  64
#define ROWS (S_ * N_)          // 200000, divisible by 16
#define RTILES (ROWS / 16)      // 12500

typedef __attribute__((ext_vector_type(2))) float v2f;
typedef __attribute__((ext_vector_type(8))) float v8f;

__device__ __forceinline__ float celu1(float x) {
    return x > 0.0f ? x : (__expf(x) - 1.0f);
}

// ---------------------------------------------------------------------------
// Kernel 1: h = celu(z @ W + b) for both W_cur and W_nbr, via f32 WMMA.
// One wave (32 lanes) computes a 16-row x 64-col tile for one of the two MLPs.
// The "nbr" wave also zero-initializes the matching agg tile (ws not re-poisoned
// between graph replays, so we must re-zero every launch).
// ---------------------------------------------------------------------------
__global__ __launch_bounds__(128) void mlp_wmma_kernel(
    const float* __restrict__ z,
    const float* __restrict__ Wc, const float* __restrict__ bc,
    const float* __restrict__ Wn, const float* __restrict__ bn,
    float* __restrict__ h_cur, float* __restrict__ h_nbr, float* __restrict__ agg)
{
    const int lane = threadIdx.x & 31;
    const int gw   = blockIdx.x * 4 + (threadIdx.x >> 5);   // global wave id
    const int tile = gw >> 1;                               // 16-row tile
    const int which = gw & 1;                               // 0 = cur, 1 = nbr
    if (tile >= RTILES) return;

    const float* __restrict__ W = which ? Wn : Wc;
    const float* __restrict__ b = which ? bn : bc;
    float* __restrict__ out = which ? h_nbr : h_cur;

    const int r0 = tile * 16;
    const int m  = lane & 15;       // row within tile (A) / col within n-tile (B)
    const int hi = lane >> 4;       // lane-half: K offset of +2 (A/B), M offset of +8 (C/D)

    v8f acc0 = {}, acc1 = {}, acc2 = {}, acc3 = {};

    #pragma unroll
    for (int kk = 0; kk < 8; ++kk) {            // K = 32 in steps of 4
        const int kbase = kk * 4 + hi * 2;
        v2f a;
        a.x = z[(r0 + m) * DZ_ + kbase];
        a.y = z[(r0 + m) * DZ_ + kbase + 1];

        v2f bf0, bf1, bf2, bf3;
        bf0.x = W[kbase * DH_ + ( 0 + m)];  bf0.y = W[(kbase + 1) * DH_ + ( 0 + m)];
        bf1.x = W[kbase * DH_ + (16 + m)];  bf1.y = W[(kbase + 1) * DH_ + (16 + m)];
        bf2.x = W[kbase * DH_ + (32 + m)];  bf2.y = W[(kbase + 1) * DH_ + (32 + m)];
        bf3.x = W[kbase * DH_ + (48 + m)];  bf3.y = W[(kbase + 1) * DH_ + (48 + m)];

        acc0 = __builtin_amdgcn_wmma_f32_16x16x4_f32(false, a, false, bf0, (short)0, acc0, false, false);
        acc1 = __builtin_amdgcn_wmma_f32_16x16x4_f32(false, a, false, bf1, (short)0, acc1, false, false);
        acc2 = __builtin_amdgcn_wmma_f32_16x16x4_f32(false, a, false, bf2, (short)0, acc2, false, false);
        acc3 = __builtin_amdgcn_wmma_f32_16x16x4_f32(false, a, false, bf3, (short)0, acc3, false, false);
    }

    v8f* accs[4] = { &acc0, &acc1, &acc2, &acc3 };
    #pragma unroll
    for (int nt = 0; nt < 4; ++nt) {
        const int n = nt * 16 + m;
        const float bias = b[n];
        #pragma unroll
        for (int v = 0; v < 8; ++v) {
            const int row = r0 + v + hi * 8;     // C/D layout: lanes 16-31 hold M+8
            float x = (*accs[nt])[v] + bias;
            out[row * DH_ + n] = celu1(x);
            if (which) agg[row * DH_ + n] = 0.0f;
        }
    }
}

// ---------------------------------------------------------------------------
// Kernel 2: agg[s, dst, :] += h_nbr[s, src, :] for every directed edge.
// Lane -> channel (64 contiguous floats of one edge's src row): coalesced
// gather, row-contiguous f32 HW atomics into L2 (agg is L2-resident, 51 MB).
// grid = (E*64/256, S)
// ---------------------------------------------------------------------------
__global__ __launch_bounds__(256) void scatter_kernel(
    const float* __restrict__ h_nbr,
    const int*   __restrict__ edges,   // [2, E]: edges[e]=src, edges[E+e]=dst
    float* __restrict__ agg)
{
    const int idx = blockIdx.x * 256 + threadIdx.x;   // over E*64
    const int c = idx & 63;
    const int e = idx >> 6;
    const int s = blockIdx.y;
    if (e >= E_) return;

    const int src = edges[e];
    const int dst = edges[E_ + e];
    const float val = h_nbr[((size_t)s * N_ + src) * DH_ + c];
    unsafeAtomicAdd(&agg[((size_t)s * N_ + dst) * DH_ + c], val);
}

// ---------------------------------------------------------------------------
// Kernel 3: dz = concat(h_cur, agg) @ W_out + b_out, via f32 WMMA.
// One wave per 16-row tile; K = 128 (first 64 from h_cur, last 64 from agg),
// N = 32 -> two 16x16 output tiles.
// ---------------------------------------------------------------------------
__global__ __launch_bounds__(128) void out_wmma_kernel(
    const float* __restrict__ h_cur, const float* __restrict__ agg,
    const float* __restrict__ Wo, const float* __restrict__ bo,
    float* __restrict__ dz)
{
    const int lane = threadIdx.x & 31;
    const int tile = blockIdx.x * 4 + (threadIdx.x >> 5);
    if (tile >= RTILES) return;

    const int r0 = tile * 16;
    const int m  = lane & 15;
    const int hi = lane >> 4;

    v8f acc0 = {}, acc1 = {};

    #pragma unroll
    for (int kk = 0; kk < 32; ++kk) {            // K = 128 in steps of 4
        const int kbase = kk * 4 + hi * 2;       // global K index 0..127
        const float* __restrict__ buf = (kk < 16) ? h_cur : agg;
        const int kloc = kbase & 63;             // K within the selected buffer

        v2f a;
        a.x = buf[(r0 + m) * DH_ + kloc];
        a.y = buf[(r0 + m) * DH_ + kloc + 1];

        v2f bf0, bf1;
        bf0.x = Wo[kbase * DZ_ + ( 0 + m)];  bf0.y = Wo[(kbase + 1) * DZ_ + ( 0 + m)];
        bf1.x = Wo[kbase * DZ_ + (16 + m)];  bf1.y = Wo[(kbase + 1) * DZ_ + (16 + m)];

        acc0 = __builtin_amdgcn_wmma_f32_16x16x4_f32(false, a, false, bf0, (short)0, acc0, false, false);
        acc1 = __builtin_amdgcn_wmma_f32_16x16x4_f32(false, a, false, bf1, (short)0, acc1, false, false);
    }

    #pragma unroll
    for (int nt = 0; nt < 2; ++nt) {
        const int n = nt * 16 + m;
        const float bias = bo[n];
        const v8f& acc = nt ? acc1 : acc0;
        #pragma unroll
        for (int v = 0; v < 8; ++v) {
            const int row = r0 + v + hi * 8;
            dz[row * DZ_ + n] = acc[v] + bias;
        }
    }
}

// ---------------------------------------------------------------------------
extern "C" void kernel_launch(void* const* d_in, const int* in_sizes, int n_in,
                              void* d_out, int out_size, void* d_ws, size_t ws_size,
                              hipStream_t stream) {
    const float* z     = (const float*)d_in[0];   // [S,N,DZ]
    const int*   edges = (const int*)  d_in[1];   // [2,E]
    const float* Wc    = (const float*)d_in[2];   // [DZ,DH]
    const float* bc    = (const float*)d_in[3];   // [DH]
    const float* Wn    = (const float*)d_in[4];   // [DZ,DH]
    const float* bn    = (const float*)d_in[5];   // [DH]
    const float* Wo    = (const float*)d_in[6];   // [2*DH,DZ]
    const float* bo    = (const float*)d_in[7];   // [DZ]
    float* out = (float*)d_out;                   // [S,N,DZ]

    const size_t HSZ = (size_t)ROWS * DH_;        // 12.8M floats per buffer
    float* h_cur = (float*)d_ws;
    float* h_nbr = h_cur + HSZ;
    float* agg   = h_nbr + HSZ;

    // K1: 12500 tiles x 2 MLPs = 25000 waves, 4 waves/block
    mlp_wmma_kernel<<<(RTILES * 2 + 3) / 4, 128, 0, stream>>>(
        z, Wc, bc, Wn, bn, h_cur, h_nbr, agg);

    // K2: one thread per (s, edge, channel)
    scatter_kernel<<<dim3((E_ * 64) / 256, S_), 256, 0, stream>>>(h_nbr, edges, agg);

    // K3: 12500 waves, 4 waves/block
    out_wmma_kernel<<<(RTILES + 3) / 4, 128, 0, stream>>>(h_cur, agg, Wo, bo, out);
}